// MultiscaleDeformAttnAlign_6579889897614
// MI455X (gfx1250) — compile-verified
//
#include <hip/hip_runtime.h>
#include <math.h>

// Problem constants (match reference)
#define NB   8
#define DM   128
#define NH   8
#define NP   8
#define DH   16
#define HSP  64
#define WSP  64
#define LQ   (HSP * WSP)   // 4096

typedef __attribute__((ext_vector_type(2))) float v2f;
typedef __attribute__((ext_vector_type(4))) float f4;
typedef __attribute__((ext_vector_type(8))) float v8f;

// V_WMMA_F32_16X16X4_F32 : D(16x16,f32) = A(16x4,f32) x B(4x16,f32) + C
// Fragment layouts (wave32, per ISA 7.12.2):
//   A: lane 0-15 -> M = lane ; VGPR v, lane-half h -> K = 2*h + v
//   B: lane 0-15 -> N = lane&15 ; K = 2*(lane>=16) + v   (mirror of A)
//   C/D: VGPR r: lanes 0-15 -> (M=r,   N=lane)
//                lanes16-31 -> (M=r+8, N=lane-16)
__device__ __forceinline__ v8f wmma_f32x4(v2f a, v2f b, v8f c) {
  return __builtin_amdgcn_wmma_f32_16x16x4_f32(false, a, false, b, (short)0, c,
                                               false, false);
}

// ---------------------------------------------------------------------------
// Kernel 1: value projection.  vh[b, head, l, dh] = (nbr^T @ W_val + b_val)
// One wave per (b, 16-row l-tile); computes all 128 output channels.
// ---------------------------------------------------------------------------
__global__ void k_value_gemm(const float* __restrict__ nbr,
                             const float* __restrict__ Wv,
                             const float* __restrict__ bv,
                             float* __restrict__ vh) {
  const int lane = threadIdx.x;
  const int m    = lane & 15;              // A: M index / B: N index
  const int kh   = (lane >> 4) << 1;       // K offset (0 or 2)
  const int bt   = blockIdx.x;             // b * 256 + mtile
  const int b    = bt >> 8;
  const int mb   = (bt & 255) << 4;        // l base of this tile

  // A[l][c] view of nbr[b][c][l]  (stride LQ along K, contiguous along M)
  const float* A = nbr + (size_t)b * DM * LQ + (mb + m);

  v8f acc[8] = {};
  for (int ks = 0; ks < DM; ks += 4) {
    v2f a;
    a.x = A[(size_t)(ks + kh) * LQ];
    a.y = A[(size_t)(ks + kh + 1) * LQ];
#pragma unroll
    for (int nt = 0; nt < 8; ++nt) {
      const float* Bp = Wv + (size_t)(ks + kh) * DM + nt * 16 + m;
      v2f bb;
      bb.x = Bp[0];
      bb.y = Bp[DM];
      acc[nt] = wmma_f32x4(a, bb, acc[nt]);
    }
  }

  const int mrow = (lane >> 4) << 3;       // +8 for upper lane half
#pragma unroll
  for (int nt = 0; nt < 8; ++nt) {         // nt == head (16 cols == dh)
    const float bias = bv[nt * 16 + m];
    float* D = vh + (((size_t)(b * NH + nt)) * LQ + mb + mrow) * DH + m;
#pragma unroll
    for (int r = 0; r < 8; ++r)
      D[(size_t)r * DH] = acc[nt][r] + bias;
  }
}

// ---------------------------------------------------------------------------
// Kernel 2: sampling-offset + attention-logit projections from query.
//   off_pre[b][l][128] = 10*tanh(query@W_off + b_off)   (pixel-space offsets)
//   attn_pre[b][l][64] = query@W_attn + b_attn          (pre-softmax logits)
// ---------------------------------------------------------------------------
__global__ void k_offattn_gemm(const float* __restrict__ ext,
                               const float* __restrict__ Wo,
                               const float* __restrict__ bo,
                               const float* __restrict__ Wa,
                               const float* __restrict__ ba,
                               float* __restrict__ off_pre,
                               float* __restrict__ attn_pre) {
  const int lane = threadIdx.x;
  const int m    = lane & 15;
  const int kh   = (lane >> 4) << 1;
  const int bt   = blockIdx.x;
  const int b    = bt >> 8;
  const int mb   = (bt & 255) << 4;

  const float* A = ext + (size_t)b * DM * LQ + (mb + m);

  v8f acc[12] = {};
  for (int ks = 0; ks < DM; ks += 4) {
    v2f a;
    a.x = A[(size_t)(ks + kh) * LQ];
    a.y = A[(size_t)(ks + kh + 1) * LQ];
#pragma unroll
    for (int nt = 0; nt < 8; ++nt) {       // W_off: 128 cols
      const float* Bp = Wo + (size_t)(ks + kh) * 128 + nt * 16 + m;
      v2f bb; bb.x = Bp[0]; bb.y = Bp[128];
      acc[nt] = wmma_f32x4(a, bb, acc[nt]);
    }
#pragma unroll
    for (int nt = 0; nt < 4; ++nt) {       // W_attn: 64 cols
      const float* Bp = Wa + (size_t)(ks + kh) * 64 + nt * 16 + m;
      v2f bb; bb.x = Bp[0]; bb.y = Bp[64];
      acc[8 + nt] = wmma_f32x4(a, bb, acc[8 + nt]);
    }
  }

  const int mrow = (lane >> 4) << 3;
#pragma unroll
  for (int nt = 0; nt < 8; ++nt) {
    const int col = nt * 16 + m;
    const float bias = bo[col];
    float* D = off_pre + ((size_t)b * LQ + mb + mrow) * 128 + col;
#pragma unroll
    for (int r = 0; r < 8; ++r)
      D[(size_t)r * 128] = 10.0f * tanhf(acc[nt][r] + bias);
  }
#pragma unroll
  for (int nt = 0; nt < 4; ++nt) {
    const int col = nt * 16 + m;
    const float bias = ba[col];
    float* D = attn_pre + ((size_t)b * LQ + mb + mrow) * 64 + col;
#pragma unroll
    for (int r = 0; r < 8; ++r)
      D[(size_t)r * 64] = acc[8 + nt][r] + bias;
  }
}

// ---------------------------------------------------------------------------
// Kernel 3: softmax over points + bilinear gather + weighted sum.
// One thread per (b, head, l); writes mid[b][l][head*16 .. +16].
// px = x + off_x exactly (ref-point algebra cancels: loc_x*W-0.5 = x+off_x).
// ---------------------------------------------------------------------------
__global__ void k_sample(const float* __restrict__ vh,
                         const float* __restrict__ off_pre,
                         const float* __restrict__ attn_pre,
                         float* __restrict__ mid) {
  const int tid = blockIdx.x * blockDim.x + threadIdx.x;
  if (tid >= NB * NH * LQ) return;
  const int l  = tid & (LQ - 1);
  const int bh = tid >> 12;
  const int hh = bh & (NH - 1);
  const int b  = bh >> 3;

  // softmax over the 8 points of this head
  const float* ap = attn_pre + ((size_t)b * LQ + l) * (NH * NP) + hh * NP;
  float w[NP];
  float mx = -3.0e38f;
#pragma unroll
  for (int p = 0; p < NP; ++p) { w[p] = ap[p]; mx = fmaxf(mx, w[p]); }
  float s = 0.0f;
#pragma unroll
  for (int p = 0; p < NP; ++p) { w[p] = __expf(w[p] - mx); s += w[p]; }
  const float inv = 1.0f / s;

  const float* op = off_pre + ((size_t)b * LQ + l) * (NH * NP * 2) + hh * (NP * 2);
  const float* V  = vh + (size_t)(b * NH + hh) * LQ * DH;
  const float xf = (float)(l & (WSP - 1));
  const float yf = (float)(l >> 6);

  f4 a0 = {}, a1 = {}, a2 = {}, a3 = {};
#pragma unroll
  for (int p = 0; p < NP; ++p) {
    const float wp = w[p] * inv;
    const float px = xf + op[2 * p];
    const float py = yf + op[2 * p + 1];
    const float x0f = floorf(px), y0f = floorf(py);
    const float fx = px - x0f, fy = py - y0f;
    const int x0 = (int)x0f, y0 = (int)y0f;
    const float cw[4] = { wp * (1.f - fx) * (1.f - fy),
                          wp * fx * (1.f - fy),
                          wp * (1.f - fx) * fy,
                          wp * fx * fy };
#pragma unroll
    for (int c = 0; c < 4; ++c) {
      const int xi = x0 + (c & 1);
      const int yi = y0 + (c >> 1);
      if (xi >= 0 && xi < WSP && yi >= 0 && yi < HSP) {
        const f4* sp = (const f4*)(V + ((size_t)(yi * WSP + xi)) * DH);
        const float wt = cw[c];
        a0 += wt * sp[0];
        a1 += wt * sp[1];
        a2 += wt * sp[2];
        a3 += wt * sp[3];
      }
    }
  }
  f4* dp = (f4*)(mid + ((size_t)b * LQ + l) * DM + hh * DH);
  dp[0] = a0; dp[1] = a1; dp[2] = a2; dp[3] = a3;
}

// ---------------------------------------------------------------------------
// Kernel 4: output projection, computed TRANSPOSED so stores are contiguous
// along l:  D[c'][l] = sum_c W_out[c][c'] * mid[l][c] + b_out[c']
// One wave per (b, 16-col l-tile); covers all 128 output channels.
// ---------------------------------------------------------------------------
__global__ void k_out_gemm(const float* __restrict__ mid,
                           const float* __restrict__ Wo,
                           const float* __restrict__ bo,
                           float* __restrict__ out) {
  const int lane = threadIdx.x;
  const int m    = lane & 15;
  const int kh   = (lane >> 4) << 1;
  const int bt   = blockIdx.x;
  const int b    = bt >> 8;
  const int nb_l = (bt & 255) << 4;        // l base (N dimension)

  // B[k][n] = mid[b][(nb_l+n)*128 + k]
  const float* Bb = mid + ((size_t)b * LQ + nb_l + m) * DM;

  v8f acc[8] = {};
  for (int ks = 0; ks < DM; ks += 4) {
    v2f bb;
    bb.x = Bb[ks + kh];
    bb.y = Bb[ks + kh + 1];
#pragma unroll
    for (int mt = 0; mt < 8; ++mt) {       // A[m][k] = W_out[k*128 + c']
      const float* Ap = Wo + (size_t)(ks + kh) * DM + mt * 16 + m;
      v2f a; a.x = Ap[0]; a.y = Ap[DM];
      acc[mt] = wmma_f32x4(a, bb, acc[mt]);
    }
  }

  const int mrow = (lane >> 4) << 3;
#pragma unroll
  for (int mt = 0; mt < 8; ++mt) {
    float* D = out + (size_t)b * DM * LQ + (size_t)(mt * 16 + mrow) * LQ + nb_l + m;
#pragma unroll
    for (int r = 0; r < 8; ++r)
      D[(size_t)r * LQ] = acc[mt][r] + bo[mt * 16 + mrow + r];
  }
}

// ---------------------------------------------------------------------------
extern "C" void kernel_launch(void* const* d_in, const int* in_sizes, int n_in,
                              void* d_out, int out_size, void* d_ws, size_t ws_size,
                              hipStream_t stream) {
  (void)in_sizes; (void)n_in; (void)out_size; (void)ws_size;
  const float* nbr  = (const float*)d_in[0];
  const float* ext  = (const float*)d_in[1];
  const float* Wv   = (const float*)d_in[2];
  const float* bv   = (const float*)d_in[3];
  const float* Woff = (const float*)d_in[4];
  const float* boff = (const float*)d_in[5];
  const float* Wat  = (const float*)d_in[6];
  const float* bat  = (const float*)d_in[7];
  const float* Wou  = (const float*)d_in[8];
  const float* bou  = (const float*)d_in[9];
  float* out = (float*)d_out;

  float* ws       = (float*)d_ws;
  float* vh       = ws;                                   // [B,NH,LQ,DH]  16 MiB
  float* off_pre  = vh + (size_t)NB * NH * LQ * DH;       // [B,LQ,128]    16 MiB
  float* attn_pre = off_pre + (size_t)NB * LQ * 128;      // [B,LQ,64]      8 MiB
  float* mid      = attn_pre + (size_t)NB * LQ * 64;      // [B,LQ,128]    16 MiB

  k_value_gemm  <<<NB * 256, 32, 0, stream>>>(nbr, Wv, bv, vh);
  k_offattn_gemm<<<NB * 256, 32, 0, stream>>>(ext, Woff, boff, Wat, bat,
                                              off_pre, attn_pre);
  k_sample      <<<(NB * NH * LQ + 255) / 256, 256, 0, stream>>>(vh, off_pre,
                                                                 attn_pre, mid);
  k_out_gemm    <<<NB * 256, 32, 0, stream>>>(mid, Wou, bou, out);
}